// LTCCell_74277164417412
// MI455X (gfx1250) — compile-verified
//
#include <hip/hip_runtime.h>
#include <hip/hip_bf16.h>
#include <stdint.h>

// LTC cell: B=4096, S=64, U=128, M=32, 6 ODE unfolds.
#define B_  4096
#define S_  64
#define U_  128
#define M_  32
#define UNFOLDS_ 6
#define EPS_ 1e-8f
#define BPB_ 8        // batches per block
#define THREADS_ 1024 // 32 wave32s

typedef unsigned int u32x4_t __attribute__((ext_vector_type(4)));
typedef int          i32x8_t __attribute__((ext_vector_type(8)));
typedef int          i32x4_t __attribute__((ext_vector_type(4)));

__device__ __forceinline__ float fast_sigmoid(float t) {
#if __has_builtin(__builtin_amdgcn_tanhf)
    // sigmoid(t) = 0.5 + 0.5*tanh(t/2): one v_tanh_f32 (gfx1250 trans op)
    return __builtin_fmaf(0.5f, __builtin_amdgcn_tanhf(t * 0.5f), 0.5f);
#else
    // 1/(1+2^(-t*log2e)): v_exp_f32 + v_rcp_f32
    float e = __builtin_amdgcn_exp2f(t * -1.44269504088896340736f);
    return __builtin_amdgcn_rcpf(1.0f + e);
#endif
}

// ---------------------------------------------------------------------------
// Prep: build interleaved float4 edge tables in workspace.
//   RP[s*U+u] = {sigma, sigma*mu, w*mask, w*mask*erev}        (U*U entries, 256KB)
//   SP[s*U+u] = same for sensory arrays                        (S*U entries, 128KB)
// ---------------------------------------------------------------------------
__global__ void ltc_prep(const float* __restrict__ sigma, const float* __restrict__ mu,
                         const float* __restrict__ w,     const float* __restrict__ erev,
                         const float* __restrict__ mask,
                         const float* __restrict__ ssig,  const float* __restrict__ smu,
                         const float* __restrict__ sw,    const float* __restrict__ serev,
                         const float* __restrict__ smask,
                         float4* __restrict__ rp, float4* __restrict__ sp) {
    int idx = blockIdx.x * blockDim.x + threadIdx.x;
    if (idx < U_ * U_) {
        float sg = sigma[idx];
        float wm = w[idx] * mask[idx];
        rp[idx] = make_float4(sg, sg * mu[idx], wm, wm * erev[idx]);
    } else if (idx < U_ * U_ + S_ * U_) {
        int j = idx - U_ * U_;
        float sg = ssig[j];
        float wm = sw[j] * smask[j];
        sp[j] = make_float4(sg, sg * smu[j], wm, wm * serev[j]);
    }
}

// ---------------------------------------------------------------------------
// Main ODE kernel: block = 8 batches x 128 units. Recurrent table in LDS
// via TDM; sensory table streamed from L2 (reused 8x per block).
// ---------------------------------------------------------------------------
__global__ __launch_bounds__(THREADS_, 1)
void ltc_main(const float* __restrict__ inputs, const float* __restrict__ state,
              const float* __restrict__ gleak,  const float* __restrict__ vleak,
              const float* __restrict__ cm,
              const float* __restrict__ input_w, const float* __restrict__ input_b,
              const float* __restrict__ output_w, const float* __restrict__ output_b,
              const float4* __restrict__ rp_g, const float4* __restrict__ sp_g,
              float* __restrict__ out_motor, float* __restrict__ out_v) {
    extern __shared__ char smem[];
    float4* RP    = (float4*)smem;                          // 16384 f4 = 256 KB @ LDS 0
    float*  lds_v = (float*)(smem + U_ * U_ * 16);          // 1024 floats
    float*  lds_x = (float*)(smem + U_ * U_ * 16 + THREADS_ * 4); // 512 floats

    const int tid = (int)threadIdx.x;
    const int u   = tid & (U_ - 1);
    const int bl  = tid >> 7;                               // 0..7
    const int b   = (int)blockIdx.x * BPB_ + bl;

    // ---- TDM: DMA 256KB recurrent edge table global -> LDS (wave 0 issues) ----
#if __has_builtin(__builtin_amdgcn_tensor_load_to_lds)
    if (tid < 32) {
        uint64_t ga = (uint64_t)(uintptr_t)rp_g;
        u32x4_t g0;
        g0[0] = 1u;                                         // count=1, user mode
        g0[1] = 0u;                                         // lds_addr = 0 (RP at LDS base)
        g0[2] = (unsigned)(ga & 0xFFFFFFFFu);               // global_addr[31:0]
        g0[3] = (unsigned)((ga >> 32) & 0x1FFFFFFu) | (2u << 30); // addr[56:32] | type=2
        i32x8_t g1;
        g1[0] = (int)(2u << 16);                            // data_size = 4B
        g1[1] = (int)(16384u << 16);                        // tensor_dim0[15:0] = 16384
        g1[2] = (int)(4u << 16);                            // dim0 hi=0 | tensor_dim1 = 4
        g1[3] = (int)(16384u << 16);                        // dim1 hi=0 | tile_dim0 = 16384
        g1[4] = 4;                                          // tile_dim1 = 4, tile_dim2 = 0
        g1[5] = 16384;                                      // tensor_dim0_stride = 16384
        g1[6] = 0;
        g1[7] = 0;
        i32x4_t gz4 = {0, 0, 0, 0};
        i32x8_t gz8 = {0, 0, 0, 0, 0, 0, 0, 0};
        // 6-arg toolchain form: (g0, g1, g2, g3, g_ext, cpol)
        __builtin_amdgcn_tensor_load_to_lds(g0, g1, gz4, gz4, gz8, 0);
#if __has_builtin(__builtin_amdgcn_s_wait_tensorcnt)
        __builtin_amdgcn_s_wait_tensorcnt(0);
#else
        asm volatile("s_wait_tensorcnt 0x0" ::: "memory");
#endif
    }
#else
    for (int i = tid; i < U_ * U_; i += THREADS_) RP[i] = rp_g[i];
#endif

    // ---- stage x = inputs*input_w + input_b, and v = state ----
    if (tid < BPB_ * S_) {
        int xb = tid >> 6, xs = tid & (S_ - 1);
        lds_x[tid] = __builtin_fmaf(inputs[((int)blockIdx.x * BPB_ + xb) * S_ + xs],
                                    input_w[xs], input_b[xs]);
    }
    lds_v[tid] = state[b * U_ + u];
    __syncthreads();

    // per-u constants
    const float gl  = gleak[u];
    const float cmt = cm[u] * (float)UNFOLDS_;
    const float lk  = gl * vleak[u];

    // ---- sensory terms (per batch, reused across unfolds) ----
    float numS = 0.0f, denS = 0.0f;
#pragma unroll 4
    for (int s = 0; s < S_; ++s) {
        float  xs = lds_x[bl * S_ + s];                     // LDS broadcast
        float4 p  = sp_g[s * U_ + u];                       // L2-resident, b128
        float  sg = fast_sigmoid(__builtin_fmaf(p.x, xs, -p.y));
        numS = __builtin_fmaf(sg, p.w, numS);
        denS = __builtin_fmaf(sg, p.z, denS);
    }

    // ---- 6 fused Euler unfolds ----
    float vnew = lds_v[tid];
    for (int it = 0; it < UNFOLDS_; ++it) {
        float num = numS, den = denS;
#pragma unroll 4
        for (int s = 0; s < U_; ++s) {
            float  vs = lds_v[bl * U_ + s];                 // LDS broadcast
            float4 p  = RP[s * U_ + u];                     // ds_load_b128, stride-1 lanes
            float  sg = fast_sigmoid(__builtin_fmaf(p.x, vs, -p.y));
            num = __builtin_fmaf(sg, p.w, num);
            den = __builtin_fmaf(sg, p.z, den);
        }
        float vo = lds_v[tid];
        vnew = (__builtin_fmaf(cmt, vo, lk) + num) / (cmt + gl + den + EPS_);
        __syncthreads();
        lds_v[tid] = vnew;
        __syncthreads();
    }

    // ---- outputs: tuple (motor [B,M], v [B,U]) concatenated ----
    out_v[b * U_ + u] = vnew;
    if (u < M_)
        out_motor[b * M_ + u] = __builtin_fmaf(vnew, output_w[u], output_b[u]);
}

// ---------------------------------------------------------------------------
extern "C" void kernel_launch(void* const* d_in, const int* in_sizes, int n_in,
                              void* d_out, int out_size, void* d_ws, size_t ws_size,
                              hipStream_t stream) {
    const float* inputs   = (const float*)d_in[0];
    const float* state    = (const float*)d_in[1];
    const float* gleak    = (const float*)d_in[2];
    const float* vleak    = (const float*)d_in[3];
    const float* cm       = (const float*)d_in[4];
    const float* sigma    = (const float*)d_in[5];
    const float* mu       = (const float*)d_in[6];
    const float* w        = (const float*)d_in[7];
    const float* erev     = (const float*)d_in[8];
    const float* ssig     = (const float*)d_in[9];
    const float* smu      = (const float*)d_in[10];
    const float* sw       = (const float*)d_in[11];
    const float* serev    = (const float*)d_in[12];
    const float* mask     = (const float*)d_in[13];
    const float* smask    = (const float*)d_in[14];
    const float* input_w  = (const float*)d_in[15];
    const float* input_b  = (const float*)d_in[16];
    const float* output_w = (const float*)d_in[17];
    const float* output_b = (const float*)d_in[18];

    float4* rp = (float4*)d_ws;               // 16384 float4 (256 KB)
    float4* sp = rp + U_ * U_;                // 8192 float4 (128 KB)

    float* out_motor = (float*)d_out;         // [B, M]
    float* out_v     = out_motor + B_ * M_;   // [B, U]

    int tot = U_ * U_ + S_ * U_;
    ltc_prep<<<(tot + 255) / 256, 256, 0, stream>>>(
        sigma, mu, w, erev, mask, ssig, smu, sw, serev, smask, rp, sp);

    size_t shmem = (size_t)U_ * U_ * 16 + THREADS_ * 4 + BPB_ * S_ * 4; // 256KB + 4KB + 2KB
    (void)hipFuncSetAttribute((const void*)ltc_main,
                              hipFuncAttributeMaxDynamicSharedMemorySize, (int)shmem);
    ltc_main<<<B_ / BPB_, THREADS_, shmem, stream>>>(
        inputs, state, gleak, vleak, cm, input_w, input_b, output_w, output_b,
        rp, sp, out_motor, out_v);
}